// ProcessNeurons_50517405335592
// MI455X (gfx1250) — compile-verified
//
#include <hip/hip_runtime.h>
#include <math.h>

// Problem constants (match reference)
constexpr int B_    = 2;
constexpr int S_    = 1024;
constexpr int H_    = 1024;
constexpr int NIN_  = 512;
constexpr int NPROC = 32;
constexpr int KSPLIT = 8;              // split of S for the contexts GEMM
constexpr int SCHUNK = S_ / KSPLIT;    // 128

typedef __attribute__((ext_vector_type(2))) float v2f;
typedef __attribute__((ext_vector_type(8))) float v8f;

__device__ __forceinline__ float wave_reduce_add(float v) {
  #pragma unroll
  for (int off = 16; off > 0; off >>= 1) v += __shfl_xor(v, off, 32);
  return v;
}

// ---------------------------------------------------------------------------
// K1: inverse L2 norm of each pattern template row (32 rows x 512)
// ---------------------------------------------------------------------------
__global__ __launch_bounds__(256)
void k_tmpl_invnorm(const float* __restrict__ tmpl, float* __restrict__ tinv) {
  const int n = blockIdx.x;
  const int t = threadIdx.x;
  float x0 = tmpl[n * NIN_ + t];
  float x1 = tmpl[n * NIN_ + t + 256];
  float p  = x0 * x0 + x1 * x1;
  __shared__ float red[8];
  p = wave_reduce_add(p);
  if ((t & 31) == 0) red[t >> 5] = p;
  __syncthreads();
  if (t == 0) {
    float s = 0.f;
    #pragma unroll
    for (int i = 0; i < 8; ++i) s += red[i];
    tinv[n] = 1.0f / fmaxf(sqrtf(s), 1e-12f);
  }
}

// ---------------------------------------------------------------------------
// K2: router: P[b,s,n] = sigmoid( <act_norm[b,s,:], tmpl_norm[n,:]> )
// One block per token; activation row cached in LDS; 8 waves x 4 templates.
// P written directly to its final home (tail of d_out).
// ---------------------------------------------------------------------------
__global__ __launch_bounds__(256)
void k_router(const float* __restrict__ act, const float* __restrict__ tmpl,
              const float* __restrict__ tinv, float* __restrict__ P) {
  const int bs = blockIdx.x;          // 0..B*S-1
  const int t  = threadIdx.x;
  __shared__ float s_act[NIN_];
  __shared__ float red[8];
  __shared__ float s_inv;

  float x0 = act[bs * NIN_ + t];
  float x1 = act[bs * NIN_ + t + 256];
  s_act[t] = x0;
  s_act[t + 256] = x1;
  float p = x0 * x0 + x1 * x1;
  p = wave_reduce_add(p);
  if ((t & 31) == 0) red[t >> 5] = p;
  __syncthreads();
  if (t == 0) {
    float s = 0.f;
    #pragma unroll
    for (int i = 0; i < 8; ++i) s += red[i];
    s_inv = 1.0f / fmaxf(sqrtf(s), 1e-12f);
  }
  __syncthreads();

  const int wave = t >> 5;
  const int lane = t & 31;
  #pragma unroll
  for (int j = 0; j < 4; ++j) {
    const int n = wave * 4 + j;       // 0..31
    const float* tr = tmpl + n * NIN_;
    float d = 0.f;
    #pragma unroll 4
    for (int a = lane; a < NIN_; a += 32) d = fmaf(s_act[a], tr[a], d);
    d = wave_reduce_add(d);
    if (lane == 0) {
      float z = d * s_inv * tinv[n];
      P[bs * NPROC + n] = 1.0f / (1.0f + expf(-z));
    }
  }
}

// ---------------------------------------------------------------------------
// K3: psum[b,n] = sum_s P[b,s,n]
// ---------------------------------------------------------------------------
__global__ __launch_bounds__(256)
void k_psum(const float* __restrict__ P, float* __restrict__ psum) {
  const int b = blockIdx.x >> 5;
  const int n = blockIdx.x & 31;
  const int t = threadIdx.x;
  float s = 0.f;
  for (int ss = t; ss < S_; ss += 256) s += P[(b * S_ + ss) * NPROC + n];
  __shared__ float red[8];
  s = wave_reduce_add(s);
  if ((t & 31) == 0) red[t >> 5] = s;
  __syncthreads();
  if (t == 0) {
    float tot = 0.f;
    #pragma unroll
    for (int i = 0; i < 8; ++i) tot += red[i];
    psum[b * NPROC + n] = tot;
  }
}

// ---------------------------------------------------------------------------
// K4: partial contexts via WMMA f32 16x16x4, K(S) split 8-way for latency
// hiding (2048 waves instead of 256). Partial 16x16 tiles are accumulated
// into ctx with native f32 global atomics (zeroed by memset beforehand).
//  A (16x4):  lane L holds A[M=L%16, K=kb..kb+1],  kb = (L<16 ? 0 : 2)
//  B (4x16):  lane L holds B[K=kb..kb+1, N=L%16]
//  D (16x16): VGPR r, lanes 0-15 -> M=r, lanes 16-31 -> M=r+8, N=lane%16
// ---------------------------------------------------------------------------
__global__ __launch_bounds__(256)
void k_contexts_wmma(const float* __restrict__ X, const float* __restrict__ P,
                     float* __restrict__ ctx) {
  const int wave = (blockIdx.x * blockDim.x + threadIdx.x) >> 5; // 0..2047
  const int lane = threadIdx.x & 31;
  const int ht = wave & 63;
  const int nt = (wave >> 6) & 1;
  const int sc = (wave >> 7) & (KSPLIT - 1);
  const int b  = wave >> 10;
  const int n0 = nt * 16, h0 = ht * 16;
  const int m  = lane & 15;
  const int kb = (lane >> 4) << 1;            // 0 or 2

  const float* Pb = P + b * (S_ * NPROC);
  const float* Xb = X + (size_t)b * (S_ * H_);
  const int sbeg = sc * SCHUNK;
  v8f c = {};
  #pragma unroll 8
  for (int s0 = sbeg; s0 < sbeg + SCHUNK; s0 += 4) {
    v2f a, bb;
    a.x  = Pb[(s0 + kb) * NPROC + n0 + m];
    a.y  = Pb[(s0 + kb + 1) * NPROC + n0 + m];
    bb.x = Xb[(s0 + kb) * H_ + h0 + m];
    bb.y = Xb[(s0 + kb + 1) * H_ + h0 + m];
    c = __builtin_amdgcn_wmma_f32_16x16x4_f32(false, a, false, bb,
                                              (short)0, c, false, false);
  }
  #pragma unroll
  for (int r = 0; r < 8; ++r) {
    const int n = n0 + r + ((lane < 16) ? 0 : 8);
    unsafeAtomicAdd(&ctx[((b * NPROC) + n) * H_ + h0 + m], c[r]);
  }
}

// ---------------------------------------------------------------------------
// K4b: normalize contexts in place: ctx /= (psum + 1e-8)
// ---------------------------------------------------------------------------
__global__ __launch_bounds__(256)
void k_ctx_norm(float* __restrict__ ctx, const float* __restrict__ psum) {
  const int idx = blockIdx.x * blockDim.x + threadIdx.x; // 0..65535
  const int n = (idx >> 10) & (NPROC - 1);
  const int b = idx >> 15;
  ctx[idx] = ctx[idx] / (psum[b * NPROC + n] + 1e-8f);
}

// ---------------------------------------------------------------------------
// K5: transformed[b,n,k] = <ctx[b,n,:], W[n*H+k,:]>  (the 128MB streaming step)
// One wave per (n,k); W row read coalesced as float4, touched exactly once,
// both batches in a single pass. This is the bandwidth-dominant kernel.
// ---------------------------------------------------------------------------
__global__ __launch_bounds__(256)
void k_transform_dot(const float* __restrict__ W, const float* __restrict__ ctx,
                     float* __restrict__ T) {
  const int wave = (blockIdx.x * blockDim.x + threadIdx.x) >> 5; // 0..32767
  const int lane = threadIdx.x & 31;
  const int n = wave >> 10;        // / H
  const int k = wave & (H_ - 1);   // % H
  const float4* wr = (const float4*)(W + (size_t)(n * H_ + k) * H_);
  const float4* c0 = (const float4*)(ctx + (0 * NPROC + n) * H_);
  const float4* c1 = (const float4*)(ctx + (1 * NPROC + n) * H_);
  float a0 = 0.f, a1 = 0.f;
  #pragma unroll
  for (int i = lane; i < H_ / 4; i += 32) {   // 8 iterations, b128 loads
    float4 w = wr[i];
    float4 x = c0[i];
    float4 y = c1[i];
    a0 = fmaf(w.x, x.x, fmaf(w.y, x.y, fmaf(w.z, x.z, fmaf(w.w, x.w, a0))));
    a1 = fmaf(w.x, y.x, fmaf(w.y, y.y, fmaf(w.z, y.z, fmaf(w.w, y.w, a1))));
  }
  a0 = wave_reduce_add(a0);
  a1 = wave_reduce_add(a1);
  if (lane == 0) {
    T[(0 * NPROC + n) * H_ + k] = a0;
    T[(1 * NPROC + n) * H_ + k] = a1;
  }
}

// ---------------------------------------------------------------------------
// K6: combined[b,s,h] = sum_n P[b,s,n] * T[b,n,h]
// WMMA f32 16x16x4: per batch M=1024(s) x N=1024(h), K=32(n). 8192 waves,
// 8 fully-unrolled WMMA steps with an independent load front.
// ---------------------------------------------------------------------------
__global__ __launch_bounds__(256)
void k_combine_wmma(const float* __restrict__ P, const float* __restrict__ T,
                    float* __restrict__ out) {
  const int wave = (blockIdx.x * blockDim.x + threadIdx.x) >> 5; // 0..8191
  const int lane = threadIdx.x & 31;
  const int ht = wave & 63;
  const int st = (wave >> 6) & 63;
  const int b  = wave >> 12;
  const int s0 = st * 16, h0 = ht * 16;
  const int m  = lane & 15;
  const int kb = (lane >> 4) << 1;            // 0 or 2

  const float* Pb = P + b * (S_ * NPROC);
  const float* Tb = T + b * (NPROC * H_);
  v8f c = {};
  #pragma unroll
  for (int k0 = 0; k0 < NPROC; k0 += 4) {     // 8 WMMA steps
    v2f a, bb;
    a.x  = Pb[(s0 + m) * NPROC + k0 + kb];
    a.y  = Pb[(s0 + m) * NPROC + k0 + kb + 1];
    bb.x = Tb[(k0 + kb) * H_ + h0 + m];
    bb.y = Tb[(k0 + kb + 1) * H_ + h0 + m];
    c = __builtin_amdgcn_wmma_f32_16x16x4_f32(false, a, false, bb,
                                              (short)0, c, false, false);
  }
  #pragma unroll
  for (int r = 0; r < 8; ++r) {
    const int s = s0 + r + ((lane < 16) ? 0 : 8);
    out[((size_t)b * S_ + s) * H_ + h0 + m] = c[r];
  }
}

// ---------------------------------------------------------------------------
extern "C" void kernel_launch(void* const* d_in, const int* in_sizes, int n_in,
                              void* d_out, int out_size, void* d_ws, size_t ws_size,
                              hipStream_t stream) {
  const float* X    = (const float*)d_in[0];   // intermediate      [B,S,H]
  const float* act  = (const float*)d_in[1];   // input_activations [B,S,NIN]
  const float* tmpl = (const float*)d_in[2];   // pattern_templates [NPROC,NIN]
  const float* W    = (const float*)d_in[3];   // W_transform       [NPROC*H,H]

  float* out = (float*)d_out;                  // combined [B,S,H] first
  float* P   = out + (size_t)B_ * S_ * H_;     // then process_activations [B,S,NPROC]

  // workspace: tinv(32) | psum(64) | pad to 128 | ctx(64K) | transformed(64K)
  float* ws   = (float*)d_ws;
  float* tinv = ws;
  float* psum = ws + 32;
  float* ctx  = ws + 128;
  float* T    = ws + 128 + B_ * NPROC * H_;

  // zero the atomic accumulation target (graph-capture-safe memset node)
  hipMemsetAsync(ctx, 0, (size_t)B_ * NPROC * H_ * sizeof(float), stream);

  k_tmpl_invnorm <<<NPROC,               256, 0, stream>>>(tmpl, tinv);
  k_router       <<<B_ * S_,             256, 0, stream>>>(act, tmpl, tinv, P);
  k_psum         <<<B_ * NPROC,          256, 0, stream>>>(P, psum);
  k_contexts_wmma<<<256,                 256, 0, stream>>>(X, P, ctx);   // 2048 waves
  k_ctx_norm     <<<(B_*NPROC*H_)/256,   256, 0, stream>>>(ctx, psum);
  k_transform_dot<<<(NPROC * H_) / 8,    256, 0, stream>>>(W, ctx, T);   // 32768 waves
  k_combine_wmma <<<1024,                256, 0, stream>>>(P, T, out);   // 8192 waves
}